// SPDLogEig_7576322310232
// MI455X (gfx1250) — compile-verified
//
#include <hip/hip_runtime.h>
#include <math.h>

// ---------------------------------------------------------------------------
// Batched SPD matrix logarithm, one 32x32 matrix per wave32.
//   A0 = P / trace(P)            (SPD: lambda_max <= trace  =>  spec in (0,1])
//   6x coupled Newton-Schulz square roots (matmul-only, WMMA f32 16x16x4)
//   log(I+E) via 7-term Mercator series (Horner, matmul-only)
//   logm(P) = 2^6 * log(A6) + log(trace) * I
// Memory-bound: 256 MB traffic @ 23.3 TB/s ~ 11 us; compute ~0.3 TFLOP f32.
// ---------------------------------------------------------------------------

typedef float v2f __attribute__((ext_vector_type(2)));
typedef float v8f __attribute__((ext_vector_type(8)));

#define BATCH   32768
#define WPB     4              // waves (matrices) per block
#define STRIDE  33             // padded LDS row stride (odd -> bank-conflict-free)
#define MATF    (32 * STRIDE)  // floats per LDS matrix buffer

// D = alpha*(A·B) + diag*I  for 32x32 matrices in LDS (row stride STRIDE).
// All four 16x16 C-tiles are staged in registers before any write-back, so
// dst may alias a or b (LDS ops within a wave are in-order per CDNA5 ISA).
// No __restrict__ here on purpose: the compiler must not reorder the stores
// ahead of the loads when dst aliases a source.
__device__ inline void wave_matmul(float* dst, const float* a, const float* b,
                                   float alpha, float diag, int lane)
{
    const int half = lane >> 4;   // 0: lanes 0-15, 1: lanes 16-31
    const int l15  = lane & 15;
    v8f acc[4];
#pragma unroll
    for (int t = 0; t < 4; ++t) {
        const int ti = (t >> 1) * 16;   // row tile base
        const int tj = (t & 1) * 16;    // col tile base
        v8f c = {};
#pragma unroll
        for (int kt = 0; kt < 8; ++kt) {
            const int k0 = kt * 4 + half * 2;
            // A fragment (16x4, f32): lanes 0-15 hold K=k..k+1, lanes 16-31 K=k+2..k+3
            v2f av, bv;
            av.x = a[(ti + l15) * STRIDE + k0];
            av.y = a[(ti + l15) * STRIDE + k0 + 1];
            // B fragment (4x16, f32): row striped across lanes, same K split
            bv.x = b[(k0    ) * STRIDE + tj + l15];
            bv.y = b[(k0 + 1) * STRIDE + tj + l15];
            // v_wmma_f32_16x16x4_f32: (neg_a, A, neg_b, B, c_mod, C, reuse_a, reuse_b)
            c = __builtin_amdgcn_wmma_f32_16x16x4_f32(false, av, false, bv,
                                                      (short)0, c, false, false);
        }
        acc[t] = c;
    }
    // C/D layout: VGPR r -> row (half*8 + r), lane%16 -> column
#pragma unroll
    for (int t = 0; t < 4; ++t) {
        const int ti = (t >> 1) * 16;
        const int tj = (t & 1) * 16;
#pragma unroll
        for (int r = 0; r < 8; ++r) {
            const int m = ti + half * 8 + r;
            const int n = tj + l15;
            dst[m * STRIDE + n] = alpha * acc[t][r] + ((m == n) ? diag : 0.0f);
        }
    }
}

__global__ __launch_bounds__(WPB * 32)
void spd_logm_kernel(const float* __restrict__ P, float* __restrict__ Out)
{
    __shared__ float lds[WPB * 3 * MATF];

    const int lane = threadIdx.x & 31;
    const int wv   = threadIdx.x >> 5;
    const int mat  = blockIdx.x * WPB + wv;   // grid sized exactly: no guard,
                                              // EXEC stays all-ones for WMMA
    float* Y = &lds[(wv * 3 + 0) * MATF];
    float* Z = &lds[(wv * 3 + 1) * MATF];
    float* T = &lds[(wv * 3 + 2) * MATF];

    const float* p = P + (size_t)mat * 1024;

    // ---- trace(P) via wave reduction (lane l reads P[l][l]) ----------------
    float tr = p[lane * 32 + lane];
#pragma unroll
    for (int m = 16; m >= 1; m >>= 1) tr += __shfl_xor(tr, m, 32);
    const float cinv = 1.0f / tr;

    // ---- load row `lane`, scale by 1/trace, park in LDS: Y = A0 ------------
    const float4* p4 = (const float4*)(p + lane * 32);
#pragma unroll
    for (int i = 0; i < 8; ++i) {
        float4 v = p4[i];
        Y[lane * STRIDE + 4 * i + 0] = v.x * cinv;
        Y[lane * STRIDE + 4 * i + 1] = v.y * cinv;
        Y[lane * STRIDE + 4 * i + 2] = v.z * cinv;
        Y[lane * STRIDE + 4 * i + 3] = v.w * cinv;
    }

    // ---- 6 successive square roots via coupled Newton-Schulz ---------------
#pragma unroll 1
    for (int s = 0; s < 6; ++s) {
#pragma unroll
        for (int j = 0; j < 32; ++j)
            Z[lane * STRIDE + j] = (j == lane) ? 1.0f : 0.0f;
        const int ni = (s == 0) ? 15 : ((s == 1) ? 10 : 9 - s);
#pragma unroll 1
        for (int it = 0; it < ni; ++it) {
            wave_matmul(T, Z, Y, -0.5f, 1.5f, lane);  // T = (3I - Z*Y)/2
            wave_matmul(Y, Y, T,  1.0f, 0.0f, lane);  // Y = Y*T   (-> sqrt)
            wave_matmul(Z, T, Z,  1.0f, 0.0f, lane);  // Z = T*Z   (-> inv sqrt)
        }
        // Y now holds A_{s+1} = sqrt(A_s)
    }

    // ---- log(I+E), E = A6 - I, |spec(E)| < ~0.16: 7-term Mercator/Horner ---
    Y[lane * STRIDE + lane] -= 1.0f;               // Y := E
#pragma unroll
    for (int j = 0; j < 32; ++j)                   // S6 = -1/6 I + 1/7 E
        T[lane * STRIDE + j] = (1.0f / 7.0f) * Y[lane * STRIDE + j]
                             + ((j == lane) ? (-1.0f / 6.0f) : 0.0f);
#pragma unroll 1
    for (int k = 5; k >= 1; --k) {
        const float ak = ((k & 1) ? 1.0f : -1.0f) / (float)k;  // (-1)^{k+1}/k
        wave_matmul(T, Y, T, 1.0f, ak, lane);      // S = E*S + a_k I
    }
    wave_matmul(T, Y, T, 1.0f, 0.0f, lane);        // L = E*S = log(A6)

    // ---- logm(P) = 64*L + log(trace)*I, streamed out as float4 -------------
    const float lc = logf(tr);
    float* o = Out + (size_t)mat * 1024;
    float4* o4 = (float4*)(o + lane * 32);
#pragma unroll
    for (int i = 0; i < 8; ++i) {
        float4 v;
        v.x = 64.0f * T[lane * STRIDE + 4 * i + 0] + ((4 * i + 0 == lane) ? lc : 0.0f);
        v.y = 64.0f * T[lane * STRIDE + 4 * i + 1] + ((4 * i + 1 == lane) ? lc : 0.0f);
        v.z = 64.0f * T[lane * STRIDE + 4 * i + 2] + ((4 * i + 2 == lane) ? lc : 0.0f);
        v.w = 64.0f * T[lane * STRIDE + 4 * i + 3] + ((4 * i + 3 == lane) ? lc : 0.0f);
        o4[i] = v;
    }
}

extern "C" void kernel_launch(void* const* d_in, const int* in_sizes, int n_in,
                              void* d_out, int out_size, void* d_ws, size_t ws_size,
                              hipStream_t stream)
{
    (void)in_sizes; (void)n_in; (void)out_size; (void)d_ws; (void)ws_size;
    const float* P = (const float*)d_in[0];
    float* Out     = (float*)d_out;
    spd_logm_kernel<<<BATCH / WPB, WPB * 32, 0, stream>>>(P, Out);
}